// HamiltonianSDE_39109972197642
// MI455X (gfx1250) — compile-verified
//
#include <hip/hip_runtime.h>

// ---------------------------------------------------------------------------
// HamiltonianSDE GNN drift for MI455X (gfx1250, wave32).
// All dense matmuls use V_WMMA_F32_16X16X4_F32 (fp32 matrix pipe; keeps the
// reference's fp32 precision). Edge concat operands are gathered into LDS,
// never materialized in HBM. segment_sum / grad scatters use f32 atomics.
// ---------------------------------------------------------------------------

typedef float v2f __attribute__((ext_vector_type(2)));
typedef float v8f __attribute__((ext_vector_type(8)));

#define NN 20000     // nodes
#define NE 320000    // edges
#define HID 128
#define LAY 4
#define KMSG 388     // 385 padded to multiple of 4

__device__ __forceinline__ float silu_f(float x) {
  float s = 1.0f / (1.0f + __expf(-x));
  return x * s;
}
__device__ __forceinline__ float dsilu_f(float x) {
  float s = 1.0f / (1.0f + __expf(-x));
  return s * (1.0f + x * (1.0f - s));
}

__device__ __forceinline__ v8f wmma4(v2f a, v2f b, v8f c) {
  // D = A(16x4 f32) * B(4x16 f32) + C(16x16 f32)
  return __builtin_amdgcn_wmma_f32_16x16x4_f32(false, a, false, b, (short)0, c,
                                               false, false);
}

// One 16x16 C tile per wave. A staged in LDS (16 rows, stride KP, zero-padded
// past K). B fetched per 4-K step through `bget(k,n)` (weights are L2-hot).
template <typename BF>
__device__ __forceinline__ v8f gemm_tile(const float* ldsA, int KP, int K,
                                         int n0, BF bget) {
  const int lane = threadIdx.x & 31;
  const int m = lane & 15;
  const int koff = (lane >> 4) << 1;  // 0 for lanes 0-15, 2 for lanes 16-31
  const int nn = n0 + (lane & 15);
  v8f acc = {};
  for (int k0 = 0; k0 < KP; k0 += 4) {
    v2f a, b;
    const int k = k0 + koff;
    a.x = ldsA[m * KP + k];
    a.y = ldsA[m * KP + k + 1];
    b.x = (k < K) ? bget(k, nn) : 0.0f;
    b.y = (k + 1 < K) ? bget(k + 1, nn) : 0.0f;
    acc = wmma4(a, b, acc);
  }
  return acc;
}

// ---------------- forward -------------------------------------------------

// h0 = x @ W_node + b_node       [NN,16]x[16,128]
__global__ void k_node_embed(const float* __restrict__ x,
                             const float* __restrict__ W,
                             const float* __restrict__ b,
                             float* __restrict__ h) {
  __shared__ float ldsA[16 * 16];
  const int row0 = blockIdx.x * 16, tid = threadIdx.x;
  for (int i = tid; i < 16 * 16; i += 256) {
    int r = i >> 4, k = i & 15;
    ldsA[i] = x[(row0 + r) * 16 + k];
  }
  __syncthreads();
  const int w = tid >> 5, lane = tid & 31, n0 = w * 16;
  auto bget = [&](int k, int n) { return W[k * HID + n]; };
  v8f acc = gemm_tile(ldsA, 16, 16, n0, bget);
  const int col = n0 + (lane & 15);
  const float bias = b[col];
  for (int j = 0; j < 8; ++j) {
    int r = row0 + j + ((lane >> 4) << 3);
    h[r * HID + col] = acc[j] + bias;
  }
}

// e0 = concat(edge_attr[:,:3], edge_attr[:,3:] @ W_eemb + b_eemb)
__global__ void k_edge_embed(const float* __restrict__ ea,
                             const float* __restrict__ W,
                             const float* __restrict__ b,
                             float* __restrict__ e0) {
  __shared__ float ldsA[16 * 8];
  const int row0 = blockIdx.x * 16, tid = threadIdx.x;
  for (int i = tid; i < 16 * 8; i += 256) {
    int r = i >> 3, k = i & 7;
    ldsA[i] = (k < 5) ? ea[(row0 + r) * 8 + 3 + k] : 0.0f;
  }
  if (tid < 48) {  // passthrough first 3 dims
    int r = tid / 3, c = tid % 3;
    e0[(row0 + r) * HID + c] = ea[(row0 + r) * 8 + c];
  }
  __syncthreads();
  const int w = tid >> 5, lane = tid & 31, n0 = w * 16;
  auto bget = [&](int k, int n) { return (n < 125) ? W[k * 125 + n] : 0.0f; };
  v8f acc = gemm_tile(ldsA, 8, 5, n0, bget);
  const int col = n0 + (lane & 15);
  if (col < 125) {
    const float bias = b[col];
    for (int j = 0; j < 8; ++j) {
      int r = row0 + j + ((lane >> 4) << 3);
      e0[r * HID + 3 + col] = acc[j] + bias;
    }
  }
}

// dist[e], dir[e] = (q[dst]-q[src])/dist — pos never changes across layers.
__global__ void k_edge_geom(const float* __restrict__ y,
                            const int* __restrict__ ei,
                            float* __restrict__ dist,
                            float* __restrict__ dir) {
  const int e = blockIdx.x * blockDim.x + threadIdx.x;
  if (e >= NE) return;
  const int s = ei[e], d = ei[NE + e];
  const float rx = y[d * 6 + 0] - y[s * 6 + 0];
  const float ry = y[d * 6 + 1] - y[s * 6 + 1];
  const float rz = y[d * 6 + 2] - y[s * 6 + 2];
  const float dd = sqrtf(rx * rx + ry * ry + rz * rz + 1e-8f);
  dist[e] = dd;
  const float inv = 1.0f / dd;
  dir[e * 3 + 0] = rx * inv;
  dir[e * 3 + 1] = ry * inv;
  dir[e * 3 + 2] = rz * inv;
}

// z_m = concat(h[src],h[dst],e,dist) @ W_msg + b; agg[dst] += silu(z_m)
__global__ void k_msg_fwd(const float* __restrict__ h,
                          const float* __restrict__ e,
                          const float* __restrict__ dist,
                          const int* __restrict__ ei,
                          const float* __restrict__ W,
                          const float* __restrict__ b,
                          float* __restrict__ zm, float* __restrict__ agg) {
  __shared__ float ldsA[16 * KMSG];
  __shared__ int ldst[16];
  const int row0 = blockIdx.x * 16, tid = threadIdx.x;
  if (tid < 16) ldst[tid] = ei[NE + row0 + tid];
  for (int i = tid; i < 16 * KMSG; i += 256) {
    int r = i / KMSG, k = i % KMSG, ge = row0 + r;
    float v;
    if (k < 128)       v = h[ei[ge] * HID + k];
    else if (k < 256)  v = h[ei[NE + ge] * HID + (k - 128)];
    else if (k < 384)  v = e[ge * HID + (k - 256)];
    else if (k == 384) v = dist[ge];
    else               v = 0.0f;
    ldsA[i] = v;
  }
  __syncthreads();
  const int w = tid >> 5, lane = tid & 31, n0 = w * 16;
  auto bget = [&](int k, int n) { return (k < 385) ? W[k * HID + n] : 0.0f; };
  v8f acc = gemm_tile(ldsA, KMSG, 385, n0, bget);
  const int col = n0 + (lane & 15);
  const float bias = b[col];
  for (int j = 0; j < 8; ++j) {
    int rl = j + ((lane >> 4) << 3);
    float z = acc[j] + bias;
    zm[(row0 + rl) * HID + col] = z;
    atomicAdd(&agg[ldst[rl] * HID + col], silu_f(z));
  }
}

// u = concat(h, agg) @ W_upd + b    [NN,256]x[256,128]
__global__ void k_upd_fwd(const float* __restrict__ h,
                          const float* __restrict__ agg,
                          const float* __restrict__ W,
                          const float* __restrict__ b, float* __restrict__ u) {
  __shared__ float ldsA[16 * 256];
  const int row0 = blockIdx.x * 16, tid = threadIdx.x;
  for (int i = tid; i < 16 * 256; i += 256) {
    int r = i >> 8, k = i & 255, gr = row0 + r;
    ldsA[i] = (k < 128) ? h[gr * HID + k] : agg[gr * HID + (k - 128)];
  }
  __syncthreads();
  const int w = tid >> 5, lane = tid & 31, n0 = w * 16;
  auto bget = [&](int k, int n) { return W[k * HID + n]; };
  v8f acc = gemm_tile(ldsA, 256, 256, n0, bget);
  const int col = n0 + (lane & 15);
  const float bias = b[col];
  for (int j = 0; j < 8; ++j) {
    int r = row0 + j + ((lane >> 4) << 3);
    u[r * HID + col] = acc[j] + bias;
  }
}

// LN + silu + residual. One row per wave (wave32 shuffle reduction).
__global__ void k_ln_fwd(const float* __restrict__ u,
                         const float* __restrict__ g,
                         const float* __restrict__ bb,
                         const float* __restrict__ hin,
                         float* __restrict__ hout, float* __restrict__ mu_s,
                         float* __restrict__ rstd_s) {
  const int w = (blockIdx.x * blockDim.x + threadIdx.x) >> 5;
  const int lane = threadIdx.x & 31;
  if (w >= NN) return;
  float vals[4], s = 0.0f, sq = 0.0f;
  for (int i = 0; i < 4; ++i) {
    float v = u[w * HID + lane + 32 * i];
    vals[i] = v; s += v; sq += v * v;
  }
  for (int off = 16; off > 0; off >>= 1) {
    s += __shfl_xor(s, off, 32);
    sq += __shfl_xor(sq, off, 32);
  }
  const float mu = s * (1.0f / 128.0f);
  const float var = sq * (1.0f / 128.0f) - mu * mu;
  const float rstd = rsqrtf(var + 1e-5f);
  if (lane == 0) { mu_s[w] = mu; rstd_s[w] = rstd; }
  for (int i = 0; i < 4; ++i) {
    int c = lane + 32 * i;
    float uln = (vals[i] - mu) * rstd * g[c] + bb[c];
    hout[w * HID + c] = hin[w * HID + c] + silu_f(uln);
  }
}

// z_e = concat(h[src],h[dst],e) @ W_eu + b ;  e_out = e + silu(z_e)
__global__ void k_eu_fwd(const float* __restrict__ h,
                         const float* __restrict__ e,
                         const int* __restrict__ ei,
                         const float* __restrict__ W,
                         const float* __restrict__ b, float* __restrict__ ze,
                         float* __restrict__ eout) {
  __shared__ float ldsA[16 * 384];
  const int row0 = blockIdx.x * 16, tid = threadIdx.x;
  for (int i = tid; i < 16 * 384; i += 256) {
    int r = i / 384, k = i % 384, ge = row0 + r;
    float v;
    if (k < 128)      v = h[ei[ge] * HID + k];
    else if (k < 256) v = h[ei[NE + ge] * HID + (k - 128)];
    else              v = e[ge * HID + (k - 256)];
    ldsA[i] = v;
  }
  __syncthreads();
  const int w = tid >> 5, lane = tid & 31, n0 = w * 16;
  auto bget = [&](int k, int n) { return W[k * HID + n]; };
  v8f acc = gemm_tile(ldsA, 384, 384, n0, bget);
  const int col = n0 + (lane & 15);
  const float bias = b[col];
  for (int j = 0; j < 8; ++j) {
    int r = row0 + j + ((lane >> 4) << 3);
    float z = acc[j] + bias;
    ze[r * HID + col] = z;
    eout[r * HID + col] = e[r * HID + col] + silu_f(z);
  }
}

// ---------------- head fwd+bwd fused: dh = dV/dh4 -------------------------
// dV/dv == 1 per node, so v itself is never needed.
__global__ void k_head_grad(const float* __restrict__ h,
                            const float* __restrict__ W_out,
                            const float* __restrict__ b_out,
                            const float* __restrict__ Wh1,
                            const float* __restrict__ bh1,
                            const float* __restrict__ Wh2,
                            float* __restrict__ dh) {
  __shared__ float sh[8][128];
  __shared__ float sf[8][64];
  __shared__ float st[8][32];
  __shared__ float sdf[8][64];
  const int w = threadIdx.x >> 5, lane = threadIdx.x & 31;
  const int row = blockIdx.x * 8 + w;
  for (int i = 0; i < 4; ++i) sh[w][lane + 32 * i] = h[row * HID + lane + 32 * i];
  __syncthreads();
  for (int t = 0; t < 2; ++t) {  // f = h@W_out + b_out
    int c = lane + 32 * t;
    float acc = b_out[c];
    for (int k = 0; k < 128; ++k) acc += sh[w][k] * W_out[k * 64 + c];
    sf[w][c] = acc;
  }
  __syncthreads();
  {  // t_j = Wh2[j] * silu'(z1_j)
    float acc = bh1[lane];
    for (int c = 0; c < 64; ++c) acc += sf[w][c] * Wh1[c * 32 + lane];
    st[w][lane] = Wh2[lane] * dsilu_f(acc);
  }
  __syncthreads();
  for (int t = 0; t < 2; ++t) {  // df = t @ Wh1^T
    int c = lane + 32 * t;
    float acc = 0.0f;
    for (int j = 0; j < 32; ++j) acc += Wh1[c * 32 + j] * st[w][j];
    sdf[w][c] = acc;
  }
  __syncthreads();
  for (int i = 0; i < 4; ++i) {  // dh = df @ W_out^T
    int k = lane + 32 * i;
    float acc = 0.0f;
    for (int c = 0; c < 64; ++c) acc += W_out[k * 64 + c] * sdf[w][c];
    dh[row * HID + k] = acc;
  }
}

// ---------------- backward ------------------------------------------------

__global__ void k_dsilu_mul(const float* __restrict__ g,
                            const float* __restrict__ z,
                            float* __restrict__ dz, int n) {
  const int i = blockIdx.x * blockDim.x + threadIdx.x;
  if (i < n) dz[i] = g[i] * dsilu_f(z[i]);
}

// d_eu_in = dz_e @ W_eu^T; scatter: [0,128)->dh[src], [128,256)->dh[dst],
// [256,384)->de (+=, identity already in de)
__global__ void k_eu_bwd(const float* __restrict__ dze,
                         const float* __restrict__ W,
                         const int* __restrict__ ei, float* __restrict__ dh,
                         float* __restrict__ de) {
  __shared__ float ldsA[16 * 128];
  __shared__ int lsrc[16], ldst[16];
  const int row0 = blockIdx.x * 16, tid = threadIdx.x;
  if (tid < 16) { lsrc[tid] = ei[row0 + tid]; ldst[tid] = ei[NE + row0 + tid]; }
  for (int i = tid; i < 16 * 128; i += 256) {
    int r = i >> 7, k = i & 127;
    ldsA[i] = dze[(row0 + r) * HID + k];
  }
  __syncthreads();
  const int w = tid >> 5, lane = tid & 31;
  for (int t = w; t < 24; t += 8) {
    const int n0 = t * 16;
    auto bget = [&](int k, int n) { return W[n * HID + k]; };  // W_eu^T
    v8f acc = gemm_tile(ldsA, 128, 128, n0, bget);
    const int col = n0 + (lane & 15);
    for (int j = 0; j < 8; ++j) {
      int rl = j + ((lane >> 4) << 3);
      float v = acc[j];
      if (col < 128)       atomicAdd(&dh[lsrc[rl] * HID + col], v);
      else if (col < 256)  atomicAdd(&dh[ldst[rl] * HID + (col - 128)], v);
      else                 de[(row0 + rl) * HID + (col - 256)] += v;
    }
  }
}

// du = LN-backward(dh * silu'(uln)).  One row per wave.
__global__ void k_ln_bwd(const float* __restrict__ dh,
                         const float* __restrict__ u,
                         const float* __restrict__ mu_s,
                         const float* __restrict__ rstd_s,
                         const float* __restrict__ g,
                         const float* __restrict__ bb, float* __restrict__ du) {
  const int w = (blockIdx.x * blockDim.x + threadIdx.x) >> 5;
  const int lane = threadIdx.x & 31;
  if (w >= NN) return;
  const float mu = mu_s[w], rstd = rstd_s[w];
  float xh[4], dxh[4], s1 = 0.0f, s2 = 0.0f;
  for (int i = 0; i < 4; ++i) {
    int c = lane + 32 * i;
    float x = (u[w * HID + c] - mu) * rstd;
    xh[i] = x;
    float uln = x * g[c] + bb[c];
    float d = dh[w * HID + c] * dsilu_f(uln) * g[c];
    dxh[i] = d; s1 += d; s2 += d * x;
  }
  for (int off = 16; off > 0; off >>= 1) {
    s1 += __shfl_xor(s1, off, 32);
    s2 += __shfl_xor(s2, off, 32);
  }
  s1 *= (1.0f / 128.0f);
  s2 *= (1.0f / 128.0f);
  for (int i = 0; i < 4; ++i) {
    int c = lane + 32 * i;
    du[w * HID + c] = rstd * (dxh[i] - s1 - xh[i] * s2);
  }
}

// d_u_in = du @ W_upd^T; [0,128) -> dh += ; [128,256) -> dagg =
__global__ void k_upd_bwd(const float* __restrict__ du,
                          const float* __restrict__ W, float* __restrict__ dh,
                          float* __restrict__ dagg) {
  __shared__ float ldsA[16 * 128];
  const int row0 = blockIdx.x * 16, tid = threadIdx.x;
  for (int i = tid; i < 16 * 128; i += 256) {
    int r = i >> 7, k = i & 127;
    ldsA[i] = du[(row0 + r) * HID + k];
  }
  __syncthreads();
  const int w = tid >> 5, lane = tid & 31;
  for (int t = w; t < 16; t += 8) {
    const int n0 = t * 16;
    auto bget = [&](int k, int n) { return W[n * HID + k]; };  // W_upd^T
    v8f acc = gemm_tile(ldsA, 128, 128, n0, bget);
    const int col = n0 + (lane & 15);
    for (int j = 0; j < 8; ++j) {
      int r = row0 + j + ((lane >> 4) << 3);
      if (col < 128) dh[r * HID + col] += acc[j];
      else dagg[r * HID + (col - 128)] = acc[j];
    }
  }
}

// dz_m = dagg[dst] * silu'(z_m)
__global__ void k_dzm(const float* __restrict__ dagg,
                      const float* __restrict__ zm, const int* __restrict__ ei,
                      float* __restrict__ dzm) {
  const int i = blockIdx.x * blockDim.x + threadIdx.x;
  if (i >= NE * HID) return;
  const int r = i >> 7, c = i & 127;
  dzm[i] = dagg[ei[NE + r] * HID + c] * dsilu_f(zm[i]);
}

// d_m_in = dz_m @ W_msg^T; scatter: dh[src], dh[dst], de +=, col 384 -> dq
__global__ void k_msg_bwd(const float* __restrict__ dzm,
                          const float* __restrict__ W,
                          const int* __restrict__ ei,
                          const float* __restrict__ dir,
                          float* __restrict__ dh, float* __restrict__ de,
                          float* __restrict__ dq) {
  __shared__ float ldsA[16 * 128];
  __shared__ int lsrc[16], ldst[16];
  const int row0 = blockIdx.x * 16, tid = threadIdx.x;
  if (tid < 16) { lsrc[tid] = ei[row0 + tid]; ldst[tid] = ei[NE + row0 + tid]; }
  for (int i = tid; i < 16 * 128; i += 256) {
    int r = i >> 7, k = i & 127;
    ldsA[i] = dzm[(row0 + r) * HID + k];
  }
  __syncthreads();
  const int w = tid >> 5, lane = tid & 31;
  for (int t = w; t < 25; t += 8) {
    const int n0 = t * 16;
    auto bget = [&](int k, int n) { return (n < 385) ? W[n * HID + k] : 0.0f; };
    v8f acc = gemm_tile(ldsA, 128, 128, n0, bget);
    const int col = n0 + (lane & 15);
    for (int j = 0; j < 8; ++j) {
      int rl = j + ((lane >> 4) << 3);
      int r = row0 + rl;
      float v = acc[j];
      if (col < 128)      atomicAdd(&dh[lsrc[rl] * HID + col], v);
      else if (col < 256) atomicAdd(&dh[ldst[rl] * HID + (col - 128)], v);
      else if (col < 384) de[r * HID + (col - 256)] += v;
      else if (col == 384) {
        float dx = v * dir[r * 3 + 0];
        float dy = v * dir[r * 3 + 1];
        float dz = v * dir[r * 3 + 2];
        atomicAdd(&dq[ldst[rl] * 3 + 0], dx);
        atomicAdd(&dq[ldst[rl] * 3 + 1], dy);
        atomicAdd(&dq[ldst[rl] * 3 + 2], dz);
        atomicAdd(&dq[lsrc[rl] * 3 + 0], -dx);
        atomicAdd(&dq[lsrc[rl] * 3 + 1], -dy);
        atomicAdd(&dq[lsrc[rl] * 3 + 2], -dz);
      }
    }
  }
}

// out = concat(p, -grad_q)
__global__ void k_output(const float* __restrict__ y,
                         const float* __restrict__ dq, float* __restrict__ out) {
  const int i = blockIdx.x * blockDim.x + threadIdx.x;
  if (i >= NN) return;
  out[i * 6 + 0] = y[i * 6 + 3];
  out[i * 6 + 1] = y[i * 6 + 4];
  out[i * 6 + 2] = y[i * 6 + 5];
  out[i * 6 + 3] = -dq[i * 3 + 0];
  out[i * 6 + 4] = -dq[i * 3 + 1];
  out[i * 6 + 5] = -dq[i * 3 + 2];
}

// ---------------------------------------------------------------------------

extern "C" void kernel_launch(void* const* d_in, const int* in_sizes, int n_in,
                              void* d_out, int out_size, void* d_ws,
                              size_t ws_size, hipStream_t stream) {
  (void)in_sizes; (void)n_in; (void)out_size; (void)ws_size;

  const float* y      = (const float*)d_in[0];
  const float* x      = (const float*)d_in[1];
  const float* ea     = (const float*)d_in[2];
  const int*   ei     = (const int*)d_in[3];
  const float* W_node = (const float*)d_in[4];
  const float* b_node = (const float*)d_in[5];
  const float* W_eemb = (const float*)d_in[6];
  const float* b_eemb = (const float*)d_in[7];
  const float* W_msg  = (const float*)d_in[8];
  const float* b_msg  = (const float*)d_in[9];
  const float* W_upd  = (const float*)d_in[10];
  const float* b_upd  = (const float*)d_in[11];
  const float* ln_g   = (const float*)d_in[12];
  const float* ln_b   = (const float*)d_in[13];
  const float* W_eu   = (const float*)d_in[14];
  const float* b_eu   = (const float*)d_in[15];
  const float* W_out  = (const float*)d_in[16];
  const float* b_out  = (const float*)d_in[17];
  const float* Wh1    = (const float*)d_in[18];
  const float* bh1    = (const float*)d_in[19];
  const float* Wh2    = (const float*)d_in[20];
  float* out = (float*)d_out;

  // workspace carve-out (floats)
  float* ws = (float*)d_ws;
  size_t off = 0;
  auto alloc = [&](size_t n) { float* p = ws + off; off += n; return p; };
  float* h[LAY + 1]; for (int l = 0; l <= LAY; ++l) h[l] = alloc((size_t)NN * HID);
  float* eb[LAY + 1]; for (int l = 0; l <= LAY; ++l) eb[l] = alloc((size_t)NE * HID);
  float* zm[LAY]; for (int l = 0; l < LAY; ++l) zm[l] = alloc((size_t)NE * HID);
  float* ze[LAY]; for (int l = 0; l < LAY; ++l) ze[l] = alloc((size_t)NE * HID);
  float* u[LAY]; for (int l = 0; l < LAY; ++l) u[l] = alloc((size_t)NN * HID);
  float* mu[LAY]; for (int l = 0; l < LAY; ++l) mu[l] = alloc(NN);
  float* rstd[LAY]; for (int l = 0; l < LAY; ++l) rstd[l] = alloc(NN);
  float* dist = alloc(NE);
  float* dir  = alloc((size_t)NE * 3);
  float* agg  = alloc((size_t)NN * HID);
  float* dagg = alloc((size_t)NN * HID);
  float* du   = alloc((size_t)NN * HID);
  float* dh   = alloc((size_t)NN * HID);
  float* de   = alloc((size_t)NE * HID);
  float* dz   = alloc((size_t)NE * HID);
  float* dq   = alloc((size_t)NN * 3);

  const dim3 B256(256);
  const dim3 G_node(NN / 16);        // 1250
  const dim3 G_edge(NE / 16);        // 20000
  const dim3 G_rowsN(NN * 32 / 256); // 2500 (one row per wave)
  const dim3 G_head(NN / 8);         // 2500
  const dim3 G_ew(NE * HID / 256);   // 160000

  hipMemsetAsync(dq, 0, (size_t)NN * 3 * sizeof(float), stream);
  hipMemsetAsync(de, 0, (size_t)NE * HID * sizeof(float), stream);

  // ---------------- forward ----------------
  k_node_embed<<<G_node, B256, 0, stream>>>(x, W_node, b_node, h[0]);
  k_edge_embed<<<G_edge, B256, 0, stream>>>(ea, W_eemb, b_eemb, eb[0]);
  k_edge_geom<<<dim3((NE + 255) / 256), B256, 0, stream>>>(y, ei, dist, dir);

  for (int l = 0; l < LAY; ++l) {
    hipMemsetAsync(agg, 0, (size_t)NN * HID * sizeof(float), stream);
    k_msg_fwd<<<G_edge, B256, 0, stream>>>(h[l], eb[l], dist, ei,
                                           W_msg + (size_t)l * 385 * HID,
                                           b_msg + l * HID, zm[l], agg);
    k_upd_fwd<<<G_node, B256, 0, stream>>>(h[l], agg,
                                           W_upd + (size_t)l * 256 * HID,
                                           b_upd + l * HID, u[l]);
    k_ln_fwd<<<G_rowsN, B256, 0, stream>>>(u[l], ln_g + l * HID,
                                           ln_b + l * HID, h[l], h[l + 1],
                                           mu[l], rstd[l]);
    k_eu_fwd<<<G_edge, B256, 0, stream>>>(h[l + 1], eb[l], ei,
                                          W_eu + (size_t)l * 384 * HID,
                                          b_eu + l * HID, ze[l], eb[l + 1]);
  }

  // ---------------- backward ----------------
  k_head_grad<<<G_head, B256, 0, stream>>>(h[LAY], W_out, b_out, Wh1, bh1, Wh2,
                                           dh);
  for (int l = LAY - 1; l >= 0; --l) {
    k_dsilu_mul<<<G_ew, B256, 0, stream>>>(de, ze[l], dz, NE * HID);
    k_eu_bwd<<<G_edge, B256, 0, stream>>>(dz, W_eu + (size_t)l * 384 * HID, ei,
                                          dh, de);
    k_ln_bwd<<<G_rowsN, B256, 0, stream>>>(dh, u[l], mu[l], rstd[l],
                                           ln_g + l * HID, ln_b + l * HID, du);
    k_upd_bwd<<<G_node, B256, 0, stream>>>(du, W_upd + (size_t)l * 256 * HID,
                                           dh, dagg);
    k_dzm<<<G_ew, B256, 0, stream>>>(dagg, zm[l], ei, dz);
    k_msg_bwd<<<G_edge, B256, 0, stream>>>(dz, W_msg + (size_t)l * 385 * HID,
                                           ei, dir, dh, de, dq);
  }

  k_output<<<dim3((NN + 255) / 256), B256, 0, stream>>>(y, dq, out);
}